// RNN_59803124629906
// MI455X (gfx1250) — compile-verified
//
#include <hip/hip_runtime.h>
#include <hip/hip_bf16.h>
#include <math.h>

typedef __attribute__((ext_vector_type(2))) float v2f;
typedef __attribute__((ext_vector_type(8))) float v8f;

#define T_LEN 512
#define H_DIM 256
#define OUT_DIM 256

// ---------------------------------------------------------------------------
// Kernel 1: XW = x(512x256) @ W_ih(256x256), fp32 WMMA (V_WMMA_F32_16X16X4_F32)
// One wave (32 threads) per 16x16 C tile. grid = (N/16, M/16).
// fp32 A layout (16x4, 2 VGPRs): lanes 0-15 hold {K0,K1}, lanes 16-31 {K2,K3}.
// fp32 B layout (4x16, 2 VGPRs): v0={K0|K2}, v1={K1|K3} striped over lanes.
// ---------------------------------------------------------------------------
__global__ __launch_bounds__(32) void gemm_x_wih(const float* __restrict__ X,
                                                 const float* __restrict__ W,
                                                 float* __restrict__ C) {
    const int n0 = blockIdx.x * 16;
    const int m0 = blockIdx.y * 16;
    const int lane = threadIdx.x;      // 0..31
    const int r    = lane & 15;        // row/col within tile
    const int half = lane >> 4;        // 0: K={0,1}, 1: K={2,3}

    v8f c = {};
    const float* arow = X + (m0 + r) * H_DIM;

    #pragma unroll 4
    for (int kk = 0; kk < H_DIM; kk += 4) {
        // A 16x4 fragment: contiguous float2 per lane
        v2f a = *(const v2f*)(arow + kk + 2 * half);
        // B 4x16 fragment: rows (kk+2*half) and (kk+2*half+1), column n0+r
        v2f b;
        b.x = W[(kk + 2 * half    ) * H_DIM + n0 + r];
        b.y = W[(kk + 2 * half + 1) * H_DIM + n0 + r];
        c = __builtin_amdgcn_wmma_f32_16x16x4_f32(
                /*neg_a=*/false, a, /*neg_b=*/false, b,
                /*c_mod=*/(short)0, c, /*reuse_a=*/false, /*reuse_b=*/false);
    }

    // C/D layout: VGPR v -> rows (v) lanes0-15, (v+8) lanes16-31; N = r
    #pragma unroll
    for (int v = 0; v < 8; ++v) {
        C[(m0 + v + 8 * half) * H_DIM + n0 + r] = c[v];
    }
}

// ---------------------------------------------------------------------------
// Kernel 2: sequential recurrence h <- tanh(XW[t] + h @ W_hh), 512 steps.
// Single workgroup, 256 threads; W_hh (256 KB) + double-buffered h (2 KB)
// live in LDS (320 KB/WGP on CDNA5). Thread j owns output column j.
//  - 4 independent accumulators break the FMA dependency chain (depth 64)
//  - double-buffered h => exactly ONE barrier per step
//  - prefetch of next step's XW row hides global latency
// ---------------------------------------------------------------------------
__global__ __launch_bounds__(256) void rnn_recurrence(const float* __restrict__ XW,
                                                      const float* __restrict__ W_hh,
                                                      float* __restrict__ h_out) {
    extern __shared__ float lds[];
    float* Wl = lds;                      // 256*256 floats
    float* hbuf0 = lds + H_DIM * H_DIM;   // 256 floats
    float* hbuf1 = hbuf0 + H_DIM;         // 256 floats

    const int j = threadIdx.x;            // 0..255

    // Cooperative load of W_hh into LDS (coalesced, 256 floats per thread)
    for (int i = j; i < H_DIM * H_DIM; i += 256) Wl[i] = W_hh[i];
    hbuf0[j] = 0.0f;
    __syncthreads();

    float* h_cur = hbuf0;
    float* h_nxt = hbuf1;

    for (int t = 0; t < T_LEN; ++t) {
        float bias = XW[t * H_DIM + j];
        if (t + 1 < T_LEN) {
            // global_prefetch for next step's bias row
            __builtin_prefetch(&XW[(t + 1) * H_DIM + j], 0, 3);
        }
        float a0 = 0.0f, a1 = 0.0f, a2 = 0.0f, a3 = 0.0f;
        #pragma unroll 2
        for (int k = 0; k < H_DIM; k += 4) {
            // broadcast b128 read of h (same address all lanes: conflict-free)
            float4 hv = *(const float4*)(h_cur + k);
            // lane-consecutive reads of Wl: conflict-free banks
            a0 += hv.x * Wl[(k + 0) * H_DIM + j];
            a1 += hv.y * Wl[(k + 1) * H_DIM + j];
            a2 += hv.z * Wl[(k + 2) * H_DIM + j];
            a3 += hv.w * Wl[(k + 3) * H_DIM + j];
        }
        h_nxt[j] = tanhf(bias + (a0 + a1) + (a2 + a3));
        __syncthreads();      // writes of h_nxt done AND reads of h_cur done
        float* tmp = h_cur; h_cur = h_nxt; h_nxt = tmp;
    }
    h_out[j] = h_cur[j];
}

// ---------------------------------------------------------------------------
// Kernel 3: y = h @ fc_W + fc_b  (one block, 256 threads)
// ---------------------------------------------------------------------------
__global__ __launch_bounds__(256) void fc_gemv(const float* __restrict__ h,
                                               const float* __restrict__ fcW,
                                               const float* __restrict__ fcb,
                                               float* __restrict__ y) {
    __shared__ float hs[H_DIM];
    const int j = threadIdx.x;
    hs[j] = h[j];
    __syncthreads();
    float a0 = 0.0f, a1 = 0.0f, a2 = 0.0f, a3 = 0.0f;
    #pragma unroll 2
    for (int k = 0; k < H_DIM; k += 4) {
        float4 hv = *(const float4*)(hs + k);
        a0 += hv.x * fcW[(k + 0) * OUT_DIM + j];
        a1 += hv.y * fcW[(k + 1) * OUT_DIM + j];
        a2 += hv.z * fcW[(k + 2) * OUT_DIM + j];
        a3 += hv.w * fcW[(k + 3) * OUT_DIM + j];
    }
    y[j] = fcb[j] + (a0 + a1) + (a2 + a3);
}

// ---------------------------------------------------------------------------
// Kernel 4: broadcast y to all 512 output rows (all rows provably identical)
// ---------------------------------------------------------------------------
__global__ __launch_bounds__(256) void bcast_rows(const float* __restrict__ y,
                                                  float* __restrict__ out) {
    out[blockIdx.x * OUT_DIM + threadIdx.x] = y[threadIdx.x];
}

extern "C" void kernel_launch(void* const* d_in, const int* in_sizes, int n_in,
                              void* d_out, int out_size, void* d_ws, size_t ws_size,
                              hipStream_t stream) {
    const float* x    = (const float*)d_in[0];   // (512, 256)
    const float* W_ih = (const float*)d_in[1];   // (256, 256)
    const float* W_hh = (const float*)d_in[2];   // (256, 256)
    const float* fc_W = (const float*)d_in[3];   // (256, 256)
    const float* fc_b = (const float*)d_in[4];   // (256,)
    float* out = (float*)d_out;                  // (512, 256)

    float* XW   = (float*)d_ws;                  // 512*256 fp32
    float* hfin = XW + T_LEN * H_DIM;            // 256 fp32
    float* y    = hfin + H_DIM;                  // 256 fp32

    // 1) Parallel GEMM via fp32 WMMA
    dim3 g1(H_DIM / 16, T_LEN / 16);             // (16, 32) tiles
    gemm_x_wih<<<g1, 32, 0, stream>>>(x, W_ih, XW);

    // 2) Sequential recurrence with W_hh in 256KB LDS (double-buffered h)
    size_t smem = (size_t)(H_DIM * H_DIM + 2 * H_DIM) * sizeof(float); // ~258 KB
    rnn_recurrence<<<1, 256, smem, stream>>>(XW, W_hh, hfin);

    // 3) Final FC on the (unique) hidden row
    fc_gemv<<<1, 256, 0, stream>>>(hfin, fc_W, fc_b, y);

    // 4) Broadcast to all 512 rows
    bcast_rows<<<T_LEN, 256, 0, stream>>>(y, out);
}